// NV12ToRgb_65979287601648
// MI455X (gfx1250) — compile-verified
//
#include <hip/hip_runtime.h>

// NV12 -> RGB on gfx1250.
// Memory-bound (~302 MB @ 23.3 TB/s ≈ 13 us floor). The 3x3 color matrix +
// offset is folded into a homogeneous K=4 affine map and executed with
// V_WMMA_F32_16X16X4_F32: A = coefficient rows (R/G/B + bias col), B = 16
// pixel columns (Y,U,V,1), D rows 0..2 = RGB for 16 pixels, landing as 3
// consecutive VGPRs per lane -> contiguous 12-byte store per pixel.

typedef float v2f __attribute__((ext_vector_type(2)));
typedef float v8f __attribute__((ext_vector_type(8)));

#define FULL_W 4096
#define FULL_H 4096
#define HALF_W 2048
#define HALF_H 2048

__global__ __launch_bounds__(256) void nv12_to_rgb_wmma(
    const int* __restrict__ bytes,   // (2048,2048,6) int32, values 0..255
    const float* __restrict__ mat,   // (3,3) row-major: mat[k][c] = mat[k*3+c]
    const float* __restrict__ off,   // (3,)
    float* __restrict__ out,         // (4096,4096,3) f32
    int numWaves)
{
    const int lane = threadIdx.x & 31;
    const int wave = blockIdx.x * (blockDim.x >> 5) + (threadIdx.x >> 5);

    const int  p  = lane & 15;      // pixel column within tile / matrix row
    const bool hi = lane >= 16;     // high half-wave carries K=2,3

    // ---- Build A (16x4 f32) once per wave ----------------------------------
    // ISA A layout: lane l (m = l%16): VGPR0 = K(l<16?0:2), VGPR1 = K(l<16?1:3)
    // A[c,0..2] = mat[k][c] (Y,U,V coefficients for channel c), A[c,3] = bias.
    float ax = 0.0f, ay = 0.0f;
    if (p < 3) {
        float m0 = mat[0 * 3 + p];           // Y coeff for channel p
        float m1 = mat[1 * 3 + p];           // U coeff
        float m2 = mat[2 * 3 + p];           // V coeff
        float bias = -(off[0] * m0 + off[1] * m1 + off[2] * m2);
        ax = hi ? m2 : m0;
        ay = hi ? bias : m1;
    }
    v2f a; a.x = ax; a.y = ay;
    v8f cz = {0.f, 0.f, 0.f, 0.f, 0.f, 0.f, 0.f, 0.f};

    const int* yplane  = bytes;
    const int* uvplane = bytes + FULL_H * FULL_W;

    // One task = a 2x16 pixel tile (row pair shares its U/V samples).
    const int totalChunks = HALF_H * (FULL_W / 16);   // 2048 * 256

    for (int chunk = wave; chunk < totalChunks; chunk += numWaves) {
        const int cy = chunk >> 8;            // row-pair index (0..2047)
        const int xc = chunk & 255;           // 16-px column chunk
        const int x  = xc * 16 + p;           // pixel x (this lane's column)
        const int r0 = cy * 2;                // top row of pair

        // ---- Loads: 3 dwords/lane. Low half: Y(r0), Y(r0+1), U.
        //             High half: V (twice, same cacheline), dummy U slot. ----
        const int* uvrow = uvplane + cy * (HALF_W * 2);
        const int  uvcol = (x >> 1) * 2;

        const int* a0 = hi ? (uvrow + uvcol + 1)              // V
                           : (yplane + r0 * FULL_W + x);      // Y row0
        const int* a1 = hi ? (uvrow + uvcol + 1)              // V (reuse line)
                           : (yplane + (r0 + 1) * FULL_W + x);// Y row1
        const int* aU = uvrow + uvcol + (hi ? 1 : 0);         // U / V

        float f0 = (float)(*a0);
        float f1 = (float)(*a1);
        float fu = (float)(*aU);
        float by = hi ? 1.0f : fu;            // homogeneous 1 in K=3 row

        // ---- B operands (4x16 f32): lanes0-15 = (Y,U), lanes16-31 = (V,1) --
        v2f b0; b0.x = f0; b0.y = by;         // row r0
        v2f b1; b1.x = f1; b1.y = by;         // row r0+1

        v8f d0 = __builtin_amdgcn_wmma_f32_16x16x4_f32(
            false, a, false, b0, (short)0, cz, false, false);
        v8f d1 = __builtin_amdgcn_wmma_f32_16x16x4_f32(
            false, a, false, b1, (short)0, cz, false, false);

        // ---- D rows 0..2 = R,G,B of pixel p, in d[0..2] of lanes 0-15 ------
        if (!hi) {
            float* o0 = out + ((size_t)r0 * FULL_W + x) * 3;
            o0[0] = d0[0]; o0[1] = d0[1]; o0[2] = d0[2];
            float* o1 = out + ((size_t)(r0 + 1) * FULL_W + x) * 3;
            o1[0] = d1[0]; o1[1] = d1[1]; o1[2] = d1[2];
        }
    }
}

extern "C" void kernel_launch(void* const* d_in, const int* in_sizes, int n_in,
                              void* d_out, int out_size, void* d_ws, size_t ws_size,
                              hipStream_t stream) {
    const int*   bytes = (const int*)d_in[0];
    const float* mat   = (const float*)d_in[1];
    const float* off   = (const float*)d_in[2];
    float*       out   = (float*)d_out;

    const int blocks  = 8192;            // 8 waves/block -> 65536 waves
    const int threads = 256;
    const int numWaves = blocks * (threads / 32);   // each wave: 8 tiles

    hipLaunchKernelGGL(nv12_to_rgb_wmma, dim3(blocks), dim3(threads), 0, stream,
                       bytes, mat, off, out, numWaves);
}